// LocalNeighborhood_6777458393495
// MI455X (gfx1250) — compile-verified
//
#include <hip/hip_runtime.h>
#include <climits>

// ---- gfx1250 async-LDS builtins (probe-confirmed present on this toolchain)
// signature: (v4i addrspace(1)* global, v4i addrspace(3)* lds, imm offset, imm cpol)
typedef int v4i __attribute__((vector_size(16)));
typedef __attribute__((address_space(1))) v4i* gptr_b128;
typedef __attribute__((address_space(3))) v4i* lptr_b128;

#if __has_builtin(__builtin_amdgcn_global_load_async_to_lds_b128)
#define HAVE_ASYNC_LOAD 1
#else
#define HAVE_ASYNC_LOAD 0
#endif
#if __has_builtin(__builtin_amdgcn_global_store_async_from_lds_b128)
#define HAVE_ASYNC_STORE 1
#else
#define HAVE_ASYNC_STORE 0
#endif

static __device__ __forceinline__ void wait_async0() {
#if __has_builtin(__builtin_amdgcn_s_wait_asynccnt)
  __builtin_amdgcn_s_wait_asynccnt(0);
#else
  asm volatile("s_wait_asynccnt 0x0" ::: "memory");
#endif
}

// ---------------- problem constants ----------------
#define BB 4
#define LL 4096
#define DD 128
#define KK 16
#define TQ 64                      // queries per block
#define HALO 24                    // staged halo rows each side
#define RSTAGE (TQ + 2 * HALO)     // 112 rows staged -> 56 KB

__global__ __launch_bounds__(256)
void knn_gather_kernel(const int* __restrict__ first_index,
                       const float* __restrict__ attr,
                       float* __restrict__ out) {
  __shared__ __align__(16) float s_rows[RSTAGE * DD];  // 57344 B
  __shared__ int s_nbr[TQ * KK];                       // 4096 B

  const int blk = blockIdx.x;
  const int b   = blk / (LL / TQ);
  const int t0  = (blk % (LL / TQ)) * TQ;
  const int tid = threadIdx.x;

  const int lo = t0 - HALO;            // staged row range [lo, hi)
  const int hi = t0 + TQ + HALO;

  const float* attrB = attr + (size_t)b * LL * DD;
  const int*   cB    = first_index + b * LL;

  // ---- Phase 0: async-stage attr rows [lo,hi) (edge-clamped) into LDS ----
  for (int ch = tid; ch < RSTAGE * 32; ch += (int)blockDim.x) {
    int row = ch >> 5;                 // 32 x 16B chunks per 512B row
    int cof = (ch & 31) << 2;          // float offset in row
    int g = lo + row;
    g = (g < 0) ? 0 : (g > LL - 1 ? LL - 1 : g);
    const float* src = attrB + (size_t)g * DD + cof;
#if HAVE_ASYNC_LOAD
    __builtin_amdgcn_global_load_async_to_lds_b128(
        (gptr_b128)(void*)src, (lptr_b128)(void*)&s_rows[ch * 4], 0, 0);
#else
    *(float4*)&s_rows[ch * 4] = *(const float4*)src;
#endif
  }

  // ---- Phase 1: exact stable-(dist, j) top-16 via two-pointer merge -------
  // keys sorted ascending; f32 dist_sq is strictly monotone in |int diff|
  // here (|diff| < 2^23), so integer compares reproduce the reference order;
  // duplicate-value runs are emitted in increasing-j order (stable argsort).
  if (tid < TQ) {
    const int i  = t0 + tid;
    const int ci = cB[i];
    int k = 0;
    int l = i, r = i + 1;
    int* my = &s_nbr[tid * KK];
    while (k < KK) {
      int dl = (l >= 0) ? (ci - cB[l]) : INT_MAX;
      int dr = (r < LL) ? (cB[r] - ci) : INT_MAX;
      if (dl <= dr) {
        int v = cB[l];
        int s = l;
        while (s > 0 && cB[s - 1] == v) --s;         // left tie run [s..l]
        if (dl == dr) {                              // combined tie group
          int w = cB[r];
          int e = r;
          while (e + 1 < LL && cB[e + 1] == w) ++e;  // right tie run [r..e]
          for (int j = s; j <= l && k < KK; ++j) my[k++] = j;
          for (int j = r; j <= e && k < KK; ++j) my[k++] = j;
          l = s - 1; r = e + 1;
        } else {
          for (int j = s; j <= l && k < KK; ++j) my[k++] = j;
          l = s - 1;
        }
      } else {
        int w = cB[r];
        int e = r;
        while (e + 1 < LL && cB[e + 1] == w) ++e;
        for (int j = r; j <= e && k < KK; ++j) my[k++] = j;
        r = e + 1;
      }
    }
  }

#if HAVE_ASYNC_LOAD
  wait_async0();          // own async loads landed in LDS
#endif
  __syncthreads();        // all waves' staged rows + s_nbr visible

  // ---- Phase 2: fan out TQ*16 rows; one 512B row per wave per iteration ---
  const int wv    = tid >> 5;
  const int lane  = tid & 31;
  const int nWav  = (int)blockDim.x >> 5;
  const size_t outBase = ((size_t)b * LL + t0) * KK * DD;
  for (int row = wv; row < TQ * KK; row += nWav) {
    const int nb = s_nbr[row];                       // wave-uniform
    float* dst = out + outBase + (size_t)row * DD + lane * 4;
    if (nb >= lo && nb < hi) {
      const float* srcL = &s_rows[(nb - lo) * DD + lane * 4];
#if HAVE_ASYNC_STORE
      __builtin_amdgcn_global_store_async_from_lds_b128(
          (gptr_b128)(void*)dst, (lptr_b128)(void*)srcL, 0, 0);
#else
      *(float4*)dst = *(const float4*)srcL;
#endif
    } else {
      // rare: long duplicate runs push a neighbor outside the staged window
      // -> read straight from global (attr is 8 MB, L2-resident)
      *(float4*)dst = *(const float4*)(attrB + (size_t)nb * DD + lane * 4);
    }
  }
#if HAVE_ASYNC_STORE
  wait_async0();          // drain async stores (S_ENDPGM also waits idle)
#endif
}

extern "C" void kernel_launch(void* const* d_in, const int* in_sizes, int n_in,
                              void* d_out, int out_size, void* d_ws, size_t ws_size,
                              hipStream_t stream) {
  const int*   first_index = (const int*)d_in[0];    // (4,4096,1) int32
  const float* attr        = (const float*)d_in[1];  // (4,4096,128) f32
  float*       out         = (float*)d_out;          // (4,4096,16,128) f32
  (void)in_sizes; (void)n_in; (void)out_size; (void)d_ws; (void)ws_size;
  dim3 grid(BB * (LL / TQ));   // 256 blocks
  dim3 block(256);             // 8 wave32 waves
  knn_gather_kernel<<<grid, block, 0, stream>>>(first_index, attr, out);
}